// Channel_3212635538209
// MI455X (gfx1250) — compile-verified
//
#include <hip/hip_runtime.h>

// ---------------------------------------------------------------------------
// Channel simulation for MI455X (gfx1250, wave32).
//
// Work split:
//  * scale_kernel : dominant, bandwidth-bound elementwise complex scale.
//      Traffic ~128MB -> ~5.5us at 23.3 TB/s. float4 (b128) global loads and
//      stores; per-(n,p,s) complex factor computed once per block in LDS
//      (16 taps in parallel across the first 16 lanes). Default temporal
//      hints: the 128MB working set fits in the 192MB L2, so inputs stay
//      L2-resident across graph replays.
//  * dft_wmma_kernel : H_t = FFT(cof zero-padded to 1024). With only 16
//      nonzero taps this is H(64x1024) = cof(64x16) @ W(16x1024), done as two
//      f32 GEMMs (cos / sin twiddles) on V_WMMA_F32_16X16X4_F32, K=16 in
//      four K=4 accumulation steps. Full f32 precision, matching the
//      reference's complex64 FFT.
// ---------------------------------------------------------------------------

typedef __attribute__((ext_vector_type(2))) float v2f;
typedef __attribute__((ext_vector_type(8))) float v8f;

#define N_   16
#define P_   4
#define L_   16
#define S_   128
#define MK_  1024
#define TWO_PI 6.283185307179586f
#define INV_C  (1.0f / 3.0e8f)

// ---------------------------------------------------------------------------
// Kernel 1: out[n,p,s,m] = x[n,p,s,m] * factor[n,p,s]
// One block per (n,p,s) row: 256 threads x 2 float4 = 512 float4 = 1024
// complex elements (8KB in, 8KB out per block).
// ---------------------------------------------------------------------------
__global__ __launch_bounds__(256) void scale_kernel(
    const float4* __restrict__ x,     // (N,P,S,MK) complex as float4 pairs
    const float*  __restrict__ cof,   // (N,P,L)
    float4*       __restrict__ out) {
  __shared__ float sr[L_], si[L_];

  const int row = blockIdx.x;        // 0 .. N*P*S-1, layout (N,P,S)
  const int np  = row / S_;          // 0..63
  const int s   = row % S_;
  const int tid = threadIdx.x;

  // factor[n,p,s] = sum_l cof[l] * exp(-i * phase_coef[l] * (delay1[l] + delay2[s]))
  if (tid < L_) {
    const float cv     = cof[np * L_ + tid];                  // strictly > 0
    const float delay1 = -100.0f * logf(cv) * INV_C;
    const float delay2 = (float)s * (0.0005f / 14.0f);        // linspace step
    const float angle  = TWO_PI * (float)tid / (float)(L_ - 1);
    const float pcoef  = TWO_PI * cosf(angle) * 1000.0f;      // max_doppler=1000
    const float phase  = pcoef * (delay1 + delay2);
    float sp, cp;
    sincosf(phase, &sp, &cp);
    sr[tid] = cv * cp;      // real: cof*cos(phase)
    si[tid] = -cv * sp;     // imag: -cof*sin(phase)
  }
  __syncthreads();

  float fr = 0.0f, fi = 0.0f;
#pragma unroll
  for (int l = 0; l < L_; ++l) { fr += sr[l]; fi += si[l]; }

  const size_t base = (size_t)row * (MK_ / 2);   // 512 float4 per row
#pragma unroll
  for (int it = 0; it < 2; ++it) {
    const size_t i = base + (size_t)tid + (size_t)it * 256;
    const float4 v = x[i];                       // two complex values
    float4 o;
    o.x = v.x * fr - v.y * fi;
    o.y = v.x * fi + v.y * fr;
    o.z = v.z * fr - v.w * fi;
    o.w = v.z * fi + v.w * fr;
    out[i] = o;
  }
}

// ---------------------------------------------------------------------------
// Kernel 2: H[np,k] = sum_{l<16} cof[np,l] * exp(-2*pi*i*l*k/1024)
// One wave (32 lanes) per 16x16 output tile. grid = (Ntiles=64, Mtiles=4).
// A(16x4 f32): lanes 0-15 hold {K=k0,K=k0+1}, lanes 16-31 hold {K=k0+2,k0+3}.
// B(4x16 f32): lane's two VGPRs hold rows k0,k0+1 at column N=lane%16
//              (upper half-wave: rows k0+2,k0+3).
// D(16x16 f32): VGPR v -> M = v (lanes 0-15) / v+8 (lanes 16-31), N = lane%16.
// ---------------------------------------------------------------------------
__global__ __launch_bounds__(32) void dft_wmma_kernel(
    const float* __restrict__ cof,   // (64, 16)
    float*       __restrict__ H) {   // (64, 1024, 2) real/imag
  const int ntile = blockIdx.x;      // 0..63  (k columns)
  const int mtile = blockIdx.y;      // 0..3   (np rows)
  const int lane  = threadIdx.x;     // 0..31, full wave, EXEC all ones
  const int half  = lane >> 4;       // 0: lanes 0-15, 1: lanes 16-31
  const int l16   = lane & 15;

  const int m   = mtile * 16 + l16;  // A-matrix row (np index)
  const int col = ntile * 16 + l16;  // B/D column (frequency bin k)

  v8f acc_c = {};                    // real accumulator (cos GEMM)
  v8f acc_s = {};                    // sin GEMM (negated at store)

#pragma unroll
  for (int kc = 0; kc < 4; ++kc) {   // K=16 in four K=4 steps
    const int k0 = kc * 4 + half * 2;

    v2f a;
    a.x = cof[m * L_ + k0];
    a.y = cof[m * L_ + k0 + 1];

    // Twiddles: theta = 2*pi*((l*k) mod 1024)/1024 (mod keeps f32 accurate)
    const int t0 = (k0 * col) & (MK_ - 1);
    const int t1 = ((k0 + 1) * col) & (MK_ - 1);
    float s0, c0, s1, c1;
    sincosf((float)t0 * (TWO_PI / (float)MK_), &s0, &c0);
    sincosf((float)t1 * (TWO_PI / (float)MK_), &s1, &c1);

    v2f bc; bc.x = c0; bc.y = c1;
    v2f bs; bs.x = s0; bs.y = s1;

    // 8 args: (neg_a, A, neg_b, B, c_mod, C, reuse_a, reuse_b)
    acc_c = __builtin_amdgcn_wmma_f32_16x16x4_f32(
        false, a, false, bc, (short)0, acc_c, false, false);
    acc_s = __builtin_amdgcn_wmma_f32_16x16x4_f32(
        false, a, false, bs, (short)0, acc_s, false, false);
  }

#pragma unroll
  for (int v = 0; v < 8; ++v) {
    const int rrow = mtile * 16 + half * 8 + v;          // D-layout row
    float* dst = H + (((size_t)rrow * MK_) + col) * 2;
    dst[0] = acc_c[v];       // H.real =  sum cof*cos
    dst[1] = -acc_s[v];      // H.imag = -sum cof*sin
  }
}

// ---------------------------------------------------------------------------
extern "C" void kernel_launch(void* const* d_in, const int* in_sizes, int n_in,
                              void* d_out, int out_size, void* d_ws, size_t ws_size,
                              hipStream_t stream) {
  const float* input_ri = (const float*)d_in[0];   // (16,4,131072,2) f32
  const float* cof      = (const float*)d_in[1];   // (16,4,16) f32
  // d_in[2] = Ns (==128), baked into constants above.

  float* out_ri = (float*)d_out;                               // 16,777,216 f32
  float* H_out  = out_ri + (size_t)N_ * P_ * S_ * MK_ * 2;     // +131,072 f32

  // Elementwise complex scale: one block per (n,p,s) row.
  scale_kernel<<<N_ * P_ * S_, 256, 0, stream>>>(
      (const float4*)input_ri, cof, (float4*)out_ri);

  // 16-tap DFT as f32 WMMA GEMM: 64x1024 output in 16x16 tiles.
  dim3 dft_grid(MK_ / 16, (N_ * P_) / 16);   // (64, 4)
  dft_wmma_kernel<<<dft_grid, 32, 0, stream>>>(cof, H_out);
}